// SpikeActivation_17712445129310
// MI455X (gfx1250) — compile-verified
//
#include <hip/hip_runtime.h>
#include <hip/hip_bf16.h>

// ---------------------------------------------------------------------------
// SNN LIF scan, MI455X (gfx1250).
// Key insight: BN of an affine-transformed signal is analytic. x_t = s_t * z
// with z = x0 - mu0 and s_t a per-channel scalar recurrence, so only ONE
// (mu0, var0) reduction is needed; the 16-step time loop is barrier-free VALU.
// TDM (tensor_load_to_lds) stages the strided channel plane; the single BN
// reduction uses V_WMMA_F32_16X16X4_F32 + wave32 butterflies + LDS combine.
// ---------------------------------------------------------------------------

#define BDIM      64
#define CDIM      256
#define LDIM      512
#define TSTEPS    16
#define PLANE     (BDIM * LDIM)        // 32768 elements per channel
#define THREADS   1024                 // 32 waves (wave32)
#define ELEMS     32                   // elements per thread
#define CHUNK_ROWS 16                  // B-rows per TDM chunk
#define CHUNK_ELEMS (CHUNK_ROWS * LDIM) // 8192 (32 KB)
#define NCHUNK    (BDIM / CHUNK_ROWS)  // 4
#define BETA_LIF  0.9f
#define EPS_BN    1e-5f
#define ALPHA_SC  5.0f

typedef unsigned int uint32x4 __attribute__((ext_vector_type(4)));
typedef int          int32x4  __attribute__((ext_vector_type(4)));
typedef int          int32x8  __attribute__((ext_vector_type(8)));
typedef float        v2f      __attribute__((ext_vector_type(2)));
typedef float        v8f      __attribute__((ext_vector_type(8)));

// Stateless counter-based RNG (splitmix-style finalizer): deterministic per
// (channel, plane-index, step, draw) -> identical output on every replay.
__device__ __forceinline__ unsigned int mix32(unsigned int x) {
  x ^= x >> 16; x *= 0x7FEB352Du;
  x ^= x >> 15; x *= 0x846CA68Bu;
  x ^= x >> 16;
  return x;
}

// Dual reduction of (s, q) across a wave32.
// WMMA does the mask-16 stage for BOTH quantities in one matrix op:
//   B VGPR0 = s -> rows {0,2}; B VGPR1 = q -> rows {1,3}
//   A rows 0-7 = [1,0,1,0], rows 8-15 = [0,1,0,1]
//   D[0]: lanes 0-15 = s[n]+s[n+16], lanes 16-31 = q[n]+q[n+16]
// Remaining 4 butterfly stages stay inside each 16-lane half.
// Returns: lanes 0-15 = total(s), lanes 16-31 = total(q).
__device__ __forceinline__ float wave_reduce_pair(v2f Asel, float s, float q) {
  v2f Bv; Bv[0] = s; Bv[1] = q;
  v8f Cz = {};
  v8f D = __builtin_amdgcn_wmma_f32_16x16x4_f32(
      /*neg_a=*/false, Asel, /*neg_b=*/false, Bv,
      /*c_mod=*/(short)0, Cz, /*reuse_a=*/false, /*reuse_b=*/false);
  float v = D[0];
  #pragma unroll
  for (int m = 8; m > 0; m >>= 1) v += __shfl_xor(v, m, 32);
  return v;
}

__global__ __launch_bounds__(THREADS)
void snn_lif_tt_kernel(const float* __restrict__ act,
                       const float* __restrict__ bn_gamma,   // [T, C]
                       const float* __restrict__ threshold,  // [C, L]
                       float* __restrict__ out)              // [B, C, L]
{
  __shared__ float stage[CHUNK_ELEMS];   // 32 KB TDM landing pad
  __shared__ float redS[33];
  __shared__ float redQ[33];

  const int tid  = threadIdx.x;
  const int c    = blockIdx.x;           // channel
  const int lane = tid & 31;
  const int wv   = tid >> 5;

  // Selector A-matrix (per-lane constant).
  v2f Asel;
  Asel[0] = ((lane & 15) < 8) ? 1.0f : 0.0f;
  Asel[1] = 1.0f - Asel[0];

  // ---------------- TDM staging: global (strided 2D tile) -> LDS -> VGPRs ---
  float z[ELEMS];                        // holds x0, then z = x0 - mu0
  const unsigned long long gbase =
      (unsigned long long)(uintptr_t)(act + (size_t)c * LDIM);
  const unsigned int ldsOff = (unsigned int)(uintptr_t)(&stage[0]); // low 32b = LDS offset

  for (int k = 0; k < NCHUNK; ++k) {
    if (wv == 0) {
      unsigned long long ga =
          gbase + (unsigned long long)k * CHUNK_ROWS * (unsigned long long)(CDIM * LDIM) * 4ull;

      // D# Group 0 (ISA 08 §8.3): count=1 | lds_addr | global_addr | type=2
      uint32x4 g0;
      g0.x = 1u;
      g0.y = ldsOff;
      g0.z = (unsigned int)ga;
      g0.w = (unsigned int)((ga >> 32) & 0x01FFFFFFull) | 0x80000000u;

      // D# Group 1 (§8.4): data_size=4B, tensor 512 x 16 (stride C*L),
      // tile 512 x 16, no multicast / padding / iterate / atomic-barrier.
      int32x8 g1;
      g1[0] = 0x00020000;                        // data_size=2 (4B)
      g1[1] = (int)(512u << 16);                 // tensor_dim0 = 512
      g1[2] = (int)((unsigned)CHUNK_ROWS << 16); // tensor_dim1 = 16
      g1[3] = (int)(512u << 16);                 // tile_dim0 = 512
      g1[4] = CHUNK_ROWS;                        // tile_dim1 = 16
      g1[5] = CDIM * LDIM;                       // tensor_dim0_stride = 131072
      g1[6] = 0;
      g1[7] = 0;

      int32x4 z4 = {0, 0, 0, 0};                 // groups 2/3 unused (2D)
#if __clang_major__ >= 23
      int32x8 z8 = {0, 0, 0, 0, 0, 0, 0, 0};
      __builtin_amdgcn_tensor_load_to_lds(g0, g1, z4, z4, z8, 0);
#else
      __builtin_amdgcn_tensor_load_to_lds(g0, g1, z4, z4, 0);
#endif
      __builtin_amdgcn_s_wait_tensorcnt(0);
    }
    __syncthreads();
    #pragma unroll
    for (int j = 0; j < 8; ++j)
      z[k * 8 + j] = stage[j * THREADS + tid];   // plane idx p = k*8192 + j*1024 + tid
    __syncthreads();
  }

  // ---------------- Single (mu0, var0) reduction ----------------------------
  {
    float s = 0.0f, q = 0.0f;
    #pragma unroll
    for (int i = 0; i < ELEMS; ++i) { s += z[i]; q += z[i] * z[i]; }
    const float v = wave_reduce_pair(Asel, s, q);
    if (lane == 0)  redS[wv] = v;
    if (lane == 16) redQ[wv] = v;
    __syncthreads();
    if (wv == 0) {
      const float v2 = wave_reduce_pair(Asel, redS[lane], redQ[lane]);
      if (lane == 0)  redS[32] = v2;
      if (lane == 16) redQ[32] = v2;
    }
    __syncthreads();
  }
  const float invN = 1.0f / (float)PLANE;
  const float mu0  = redS[32] * invN;
  const float var0 = redQ[32] * invN - mu0 * mu0;

  // ---------------- Register-resident state ---------------------------------
  float mem[ELEMS];
  unsigned int accp[ELEMS / 4];          // 4 byte-counters per dword (count <= 32)
  #pragma unroll
  for (int i = 0; i < ELEMS; ++i) { z[i] -= mu0; mem[i] = 0.0f; }
  #pragma unroll
  for (int i = 0; i < ELEMS / 4; ++i) accp[i] = 0u;

  // l = p % 512 = tid % 512 for every element this thread owns.
  const float th = threshold[c * LDIM + (tid & (LDIM - 1))];
  const unsigned int gidBase = ((unsigned int)c << 15);  // c * PLANE

  // ---------------- Barrier-free time loop (all state in VGPRs) -------------
  // x_t = sc * z exactly; BN stats evolve analytically:
  //   sc <- sc * gamma_t * rsqrt(sc^2 * var0 + eps)
  float sc = 1.0f;
  #pragma unroll 1
  for (int t = 0; t < TSTEPS; ++t) {
    sc = sc * bn_gamma[t * CDIM + c] * __frsqrt_rn(sc * sc * var0 + EPS_BN);

    #pragma unroll
    for (int i = 0; i < ELEMS; ++i) {
      const float xv = sc * z[i];                      // BN-ed activation
      const float mprev = mem[i];
      const float m2 = BETA_LIF * mprev + xv - ((mprev > th) ? th : 0.0f);
      mem[i] = m2;
      unsigned int cnt = (m2 - th > 0.0f) ? 1u : 0u;   // Heaviside spike

      // second_chance_of_firing (exponential): only one branch can fire.
      const float diff = xv - th;
      const float e = __expf(-ALPHA_SC * fabsf(diff));
      const float p = (diff < 0.0f) ? e : ((diff > 0.0f) ? (1.0f - e) : 0.0f);
      const int  kk = i >> 3, jj = i & 7;
      const unsigned int pidx = (unsigned int)(kk * CHUNK_ELEMS + jj * THREADS + tid);
      const unsigned int draw = (diff < 0.0f) ? 0u : 1u;
      const unsigned int r =
          mix32(((gidBase + pidx) << 5) | ((unsigned int)t << 1) | draw);
      const float u = (float)r * 2.3283064365386963e-10f;   // [0,1)
      cnt += (u < p) ? 1u : 0u;                             // spk in {0,1,2}

      accp[i >> 2] += cnt << ((i & 3) * 8);
    }
  }

  // ---------------- Write mean over T ---------------------------------------
  const float invT = 1.0f / (float)TSTEPS;
  #pragma unroll
  for (int i = 0; i < ELEMS; ++i) {
    const int kk = i >> 3, jj = i & 7;
    const unsigned int p = (unsigned int)(kk * CHUNK_ELEMS + jj * THREADS + tid);
    const unsigned int b = p >> 9;          // p / 512
    const unsigned int l = p & 511;
    const float v = (float)((accp[i >> 2] >> ((i & 3) * 8)) & 0xFFu) * invT;
    out[((size_t)b * CDIM + c) * LDIM + l] = v;
  }
}

extern "C" void kernel_launch(void* const* d_in, const int* in_sizes, int n_in,
                              void* d_out, int out_size, void* d_ws, size_t ws_size,
                              hipStream_t stream) {
  (void)in_sizes; (void)n_in; (void)out_size; (void)d_ws; (void)ws_size;
  const float* act   = (const float*)d_in[0];  // [B, C, L]
  const float* gamma = (const float*)d_in[1];  // [T, C]
  const float* thr   = (const float*)d_in[2];  // [C, L]
  float* out = (float*)d_out;                  // [B, C, L]

  snn_lif_tt_kernel<<<dim3(CDIM), dim3(THREADS), 0, stream>>>(act, gamma, thr, out);
}